// Deconvolution1D_52432960750018
// MI455X (gfx1250) — compile-verified
//
#include <hip/hip_runtime.h>
#include <hip/hip_bf16.h>

#define B_    16
#define T_    8192
#define C_    8
#define FIL_  16
#define TPB   256
#define LDS_BYTES (4u * T_ * sizeof(float))   // re0, im0, re1, im1 = 128 KB

typedef __attribute__((ext_vector_type(2))) float v2f;
typedef __attribute__((ext_vector_type(8))) float v8f;

// ---------------------------------------------------------------------------
// One radix-16 Stockham stage over N=8192 complex values living in LDS.
// Column batch of 16 -> 16x16 tile; Y(16x16) = DFT16(16x16) * U(16x16) done as
// four chained V_WMMA_F32_16X16X4_F32 per real-accumulation chain (4 chains
// for the complex product). Twiddles are folded into U (the B operand).
//   gather : U[r][col] = in[k + m*(j + L*r)] * exp(sign*i*2*pi*j*r/(16L))
//   scatter: out[k + m*(16*j + M)] = Y[M][col]
// ---------------------------------------------------------------------------
__device__ __forceinline__ void radix16_stage_wmma(const float* __restrict__ in_re,
                                                   const float* __restrict__ in_im,
                                                   float* __restrict__ out_re,
                                                   float* __restrict__ out_im,
                                                   int L, float sign)
{
    const int   N     = T_;
    const int   m     = N / (L * 16);
    const int   lane  = threadIdx.x & 31;
    const int   wave  = threadIdx.x >> 5;      // 0..7
    const int   colN  = lane & 15;             // N index inside tile (also A's M row)
    const int   half  = lane >> 4;             // 0 or 1
    const float twoPi = 6.28318530717958647692f;

    for (int tile = wave; tile < (N / 16) / 16; tile += TPB / 32) {  // 32 tiles / 8 waves
        const int   col  = tile * 16 + colN;   // global column 0..511
        const int   k    = col % m;
        const int   j    = col / m;
        const float wang = sign * twoPi * (float)j / (float)(L * 16);
        const int   base = k + m * j;

        v8f acc_re = {0.f,0.f,0.f,0.f,0.f,0.f,0.f,0.f};
        v8f acc_im = {0.f,0.f,0.f,0.f,0.f,0.f,0.f,0.f};

        #pragma unroll
        for (int c = 0; c < 4; ++c) {
            const int K0 = 4 * c + 2 * half;   // this lane's two K slots of chunk c

            // A = DFT16 (rows M = lane&15, cols K0, K0+1), generated on the fly.
            float angA0 = sign * (twoPi / 16.0f) * (float)(colN * K0);
            float angA1 = sign * (twoPi / 16.0f) * (float)(colN * (K0 + 1));
            v2f a_re, a_im, a_imn;
            a_re.x  =  __cosf(angA0);  a_re.y  =  __cosf(angA1);
            a_im.x  =  __sinf(angA0);  a_im.y  =  __sinf(angA1);
            a_imn.x = -a_im.x;         a_imn.y = -a_im.y;   // f32 WMMA has no A-neg; do it in VALU

            // B = twiddled inputs, rows r = K0, K0+1 of this lane's column.
            v2f b_re, b_im;
            {
                const int r  = K0;
                const float xr = in_re[base + m * L * r];
                const float xi = in_im[base + m * L * r];
                const float ca = __cosf(wang * (float)r);
                const float sa = __sinf(wang * (float)r);
                b_re.x = xr * ca - xi * sa;
                b_im.x = xr * sa + xi * ca;
            }
            {
                const int r  = K0 + 1;
                const float xr = in_re[base + m * L * r];
                const float xi = in_im[base + m * L * r];
                const float ca = __cosf(wang * (float)r);
                const float sa = __sinf(wang * (float)r);
                b_re.y = xr * ca - xi * sa;
                b_im.y = xr * sa + xi * ca;
            }

            // Yr += Ar*Br - Ai*Bi ; Yi += Ar*Bi + Ai*Br
            acc_re = __builtin_amdgcn_wmma_f32_16x16x4_f32(false, a_re,  false, b_re, (short)0, acc_re, false, false);
            acc_re = __builtin_amdgcn_wmma_f32_16x16x4_f32(false, a_imn, false, b_im, (short)0, acc_re, false, false);
            acc_im = __builtin_amdgcn_wmma_f32_16x16x4_f32(false, a_re,  false, b_im, (short)0, acc_im, false, false);
            acc_im = __builtin_amdgcn_wmma_f32_16x16x4_f32(false, a_im,  false, b_re, (short)0, acc_im, false, false);
        }

        // D layout: lane holds rows M = v + 8*half of column colN (ISA 7.12.2).
        #pragma unroll
        for (int v = 0; v < 8; ++v) {
            const int rM   = v + 8 * half;
            const int oidx = k + m * (16 * j + rM);
            out_re[oidx] = acc_re[v];
            out_im[oidx] = acc_im[v];
        }
    }
}

// Final radix-2 Stockham stage (L = 4096, m = 1) — plain VALU butterflies.
__device__ __forceinline__ void radix2_stage(const float* __restrict__ in_re,
                                             const float* __restrict__ in_im,
                                             float* __restrict__ out_re,
                                             float* __restrict__ out_im,
                                             float sign)
{
    const float twoPi = 6.28318530717958647692f;
    for (int j = threadIdx.x; j < T_ / 2; j += TPB) {
        const float ur = in_re[j],          ui = in_im[j];
        const float vr = in_re[j + T_ / 2], vi = in_im[j + T_ / 2];
        const float ang = sign * twoPi * (float)j / (float)T_;
        const float c = __cosf(ang), s = __sinf(ang);
        const float tr = vr * c - vi * s;
        const float ti = vr * s + vi * c;
        out_re[2 * j]     = ur + tr;  out_im[2 * j]     = ui + ti;
        out_re[2 * j + 1] = ur - tr;  out_im[2 * j + 1] = ui - ti;
    }
}

// Full 8192-point complex FFT in LDS: radices 16,16,16,2 (Stockham autosort).
// Input & result in (re0, im0); (re1, im1) is the ping-pong buffer.
// sign = -1 forward, +1 inverse (caller applies the 1/N scale).
__device__ __forceinline__ void fft8192(float* re0, float* im0,
                                        float* re1, float* im1, float sign)
{
    radix16_stage_wmma(re0, im0, re1, im1, 1,   sign);  __syncthreads();
    radix16_stage_wmma(re1, im1, re0, im0, 16,  sign);  __syncthreads();
    radix16_stage_wmma(re0, im0, re1, im1, 256, sign);  __syncthreads();
    radix2_stage      (re1, im1, re0, im0,      sign);  __syncthreads();
}

// ---------------------------------------------------------------------------
// Kernel 1: h = FFT(w_real + i*w_imag); g = conj(h) / (|h|^2 + s)   (16 WGs)
// ---------------------------------------------------------------------------
__global__ void __launch_bounds__(TPB)
wiener_filter_kernel(const float* __restrict__ w_real, const float* __restrict__ w_imag,
                     const float* __restrict__ s,
                     float* __restrict__ g_re, float* __restrict__ g_im)
{
    extern __shared__ float smem[];
    float* re0 = smem;            float* im0 = smem + T_;
    float* re1 = smem + 2 * T_;   float* im1 = smem + 3 * T_;

    const int f = blockIdx.x;
    for (int t = threadIdx.x; t < T_; t += TPB) {
        re0[t] = w_real[f * T_ + t];
        im0[t] = w_imag[f * T_ + t];
    }
    __syncthreads();

    fft8192(re0, im0, re1, im1, -1.0f);

    const float sf = s[f];
    for (int t = threadIdx.x; t < T_; t += TPB) {
        const float hr = re0[t], hi = im0[t];
        const float inv = 1.0f / (hr * hr + hi * hi + sf);
        g_re[f * T_ + t] =  hr * inv;
        g_im[f * T_ + t] = -hi * inv;
    }
}

// ---------------------------------------------------------------------------
// Kernel 2: xf[b,c,:] = FFT(x[b,:,c])                                (128 WGs)
// ---------------------------------------------------------------------------
__global__ void __launch_bounds__(TPB)
x_fft_kernel(const float* __restrict__ x,
             float* __restrict__ xf_re, float* __restrict__ xf_im)
{
    extern __shared__ float smem[];
    float* re0 = smem;            float* im0 = smem + T_;
    float* re1 = smem + 2 * T_;   float* im1 = smem + 3 * T_;

    const int bc = blockIdx.x;            // b*C + c
    const int b  = bc / C_;
    const int c  = bc % C_;
    for (int t = threadIdx.x; t < T_; t += TPB) {
        re0[t] = x[(size_t)(b * T_ + t) * C_ + c];
        im0[t] = 0.0f;
    }
    __syncthreads();

    fft8192(re0, im0, re1, im1, -1.0f);

    for (int t = threadIdx.x; t < T_; t += TPB) {
        xf_re[(size_t)bc * T_ + t] = re0[t];
        xf_im[(size_t)bc * T_ + t] = im0[t];
    }
}

// ---------------------------------------------------------------------------
// Kernel 3: y = real(IFFT(xf * g)); out[b,t,f*C+c] = y + bias       (2048 WGs)
// ---------------------------------------------------------------------------
__global__ void __launch_bounds__(TPB)
deconv_ifft_kernel(const float* __restrict__ xf_re, const float* __restrict__ xf_im,
                   const float* __restrict__ g_re,  const float* __restrict__ g_im,
                   const float* __restrict__ bias,  float* __restrict__ out)
{
    extern __shared__ float smem[];
    float* re0 = smem;            float* im0 = smem + T_;
    float* re1 = smem + 2 * T_;   float* im1 = smem + 3 * T_;

    const int idx = blockIdx.x;                 // ((b*FIL + f)*C + c)
    const int c   = idx % C_;
    const int f   = (idx / C_) % FIL_;
    const int b   = idx / (C_ * FIL_);
    const int bc  = b * C_ + c;

    for (int t = threadIdx.x; t < T_; t += TPB) {
        const float xr = xf_re[(size_t)bc * T_ + t];
        const float xi = xf_im[(size_t)bc * T_ + t];
        const float gr = g_re[(size_t)f * T_ + t];
        const float gi = g_im[(size_t)f * T_ + t];
        re0[t] = xr * gr - xi * gi;
        im0[t] = xr * gi + xi * gr;
    }
    __syncthreads();

    fft8192(re0, im0, re1, im1, +1.0f);         // inverse transform

    const float bv    = bias[f * C_ + c];
    const float scale = 1.0f / (float)T_;
    for (int t = threadIdx.x; t < T_; t += TPB) {
        out[(size_t)(b * T_ + t) * (FIL_ * C_) + f * C_ + c] = re0[t] * scale + bv;
    }
}

// ---------------------------------------------------------------------------
// Host launcher.  d_in order: x, w_real, w_imag, s, b
// Workspace layout (floats): g_re[FIL*T] | g_im[FIL*T] | xf_re[B*C*T] | xf_im[B*C*T]
// ---------------------------------------------------------------------------
extern "C" void kernel_launch(void* const* d_in, const int* in_sizes, int n_in,
                              void* d_out, int out_size, void* d_ws, size_t ws_size,
                              hipStream_t stream)
{
    const float* x      = (const float*)d_in[0];
    const float* w_real = (const float*)d_in[1];
    const float* w_imag = (const float*)d_in[2];
    const float* s      = (const float*)d_in[3];
    const float* bias   = (const float*)d_in[4];
    float*       out    = (float*)d_out;

    float* ws    = (float*)d_ws;
    const size_t GPLANE = (size_t)FIL_ * T_;        // 131072
    const size_t XPLANE = (size_t)B_ * C_ * T_;     // 1048576
    float* g_re  = ws;
    float* g_im  = ws + GPLANE;
    float* xf_re = ws + 2 * GPLANE;
    float* xf_im = ws + 2 * GPLANE + XPLANE;

    // 128 KB of dynamic LDS per workgroup (CDNA5 WGP allows up to 320 KB).
    hipFuncSetAttribute((const void*)wiener_filter_kernel,
                        hipFuncAttributeMaxDynamicSharedMemorySize, (int)LDS_BYTES);
    hipFuncSetAttribute((const void*)x_fft_kernel,
                        hipFuncAttributeMaxDynamicSharedMemorySize, (int)LDS_BYTES);
    hipFuncSetAttribute((const void*)deconv_ifft_kernel,
                        hipFuncAttributeMaxDynamicSharedMemorySize, (int)LDS_BYTES);

    wiener_filter_kernel<<<FIL_,               TPB, LDS_BYTES, stream>>>(w_real, w_imag, s, g_re, g_im);
    x_fft_kernel        <<<B_ * C_,            TPB, LDS_BYTES, stream>>>(x, xf_re, xf_im);
    deconv_ifft_kernel  <<<B_ * FIL_ * C_,     TPB, LDS_BYTES, stream>>>(xf_re, xf_im, g_re, g_im, bias, out);
}